// StochaticPool2d_63402307224338
// MI455X (gfx1250) — compile-verified
//
#include <hip/hip_runtime.h>
#include <stdint.h>

// ---------------------------------------------------------------------------
// StochasticPool2d (eval), 2x2 stride-2, f32:  out = sum(x^2) / sum(x) per win.
// x = (B*C=planes) x 224 x 224  ->  out = planes x 112 x 112.
// Roofline: ~385 MB total traffic @ 23.3 TB/s => ~16.5 us floor; 0.5 flop/B.
// Strategy: CDNA5 async global->LDS DMA (ASYNCcnt), double-buffered per-wave
// pipeline. Each thread owns a fixed (io, jg) tile position and walks ITERS
// consecutive planes => per-stage addressing is a constant stride (no per-
// stage integer division). Division via v_rcp_f32 + 1 NR refine (<=2 ulp),
// keeping instruction issue (~10us) well under the HBM floor (~16.5us).
// ---------------------------------------------------------------------------

typedef float f4 __attribute__((ext_vector_type(4)));
typedef float f2 __attribute__((ext_vector_type(2)));
typedef int   v4i __attribute__((ext_vector_type(4)));

#define GAS __attribute__((address_space(1)))
#define LAS __attribute__((address_space(3)))

// Async copy of 16 bytes/lane from global to LDS. Tracked by ASYNCcnt.
__device__ __forceinline__ void async_ld_b128(const float* g, uint32_t lds_off) {
#if __has_builtin(__builtin_amdgcn_global_load_async_to_lds_b128)
  __builtin_amdgcn_global_load_async_to_lds_b128(
      (GAS v4i*)(uintptr_t)g, (LAS v4i*)(uintptr_t)lds_off,
      /*offset=*/0, /*cpol=*/0);
#else
  uint64_t ga = (uint64_t)(uintptr_t)g;
  asm volatile("global_load_async_to_lds_b128 %0, %1, off"
               :: "v"(lds_off), "v"(ga) : "memory");
#endif
}

// Wait until at most N async ops outstanding (per wave) + compiler mem barrier.
#define WAIT_ASYNC(n) asm volatile("s_wait_asynccnt %0" :: "n"(n) : "memory")

// n/d via v_rcp_f32 + one Newton-Raphson step: <=2 ulp, 4 VALU vs ~9 for IEEE.
__device__ __forceinline__ float fastdiv(float n, float d) {
  float r = __builtin_amdgcn_rcpf(d);
  r = __builtin_fmaf(__builtin_fmaf(-d, r, 1.0f), r, r);
  return n * r;
}

namespace {
constexpr int W     = 224;           // input width/height
constexpr int Ho    = 112;           // output rows per plane
constexpr int JG    = 56;            // pairs (2 outputs / 4 input cols) per row
constexpr int QPP   = Ho * JG;       // pairs per plane = 6272
constexpr int BLOCK = 224;           // 7 waves; 28 blocks cover one plane slice
constexpr int ITERS = 8;             // planes walked per thread (pipeline depth)
constexpr int GSTRIDE = W * W;       // per-plane input stride (floats) = 50176
constexpr int OSTRIDE = Ho * Ho;     // per-plane output stride (floats) = 12544
}

__global__ __launch_bounds__(BLOCK)
void stoch_pool2x2_async(const float* __restrict__ x, float* __restrict__ out) {
  // [double-buffer][row0/row1][BLOCK threads * 16B] = 7 KB
  __shared__ __align__(16) float lds[2][2][BLOCK * 4];

  const int tid = threadIdx.x;
  // Pair position inside a plane: fixed for all ITERS stages.
  const int q  = blockIdx.x * BLOCK + tid;   // [0, 6272)
  const int io = q / JG;                     // only division in the kernel
  const int jg = q - io * JG;
  const int plane0 = blockIdx.y * ITERS;

  const float* g0 = x + ((size_t)plane0 * W + 2 * io) * W + jg * 4; // row 2*io
  const float* g1 = g0 + W;                                          // row 2*io+1
  float* o = out + ((size_t)plane0 * Ho + io) * Ho + jg * 2;

  const uint32_t lbase = (uint32_t)(uintptr_t)&lds[0][0][0];
  uint32_t l[2][2];
#pragma unroll
  for (int b = 0; b < 2; ++b)
#pragma unroll
    for (int r = 0; r < 2; ++r)
      l[b][r] = lbase + (uint32_t)(((b * 2 + r) * BLOCK + tid) * 16);

  // Stage i loads plane (plane0+i): constant stride -> just a pointer bump.
  auto issue = [&](int i) {
    const size_t off = (size_t)i * GSTRIDE;
    async_ld_b128(g0 + off, l[i & 1][0]);
    async_ld_b128(g1 + off, l[i & 1][1]);
  };

  issue(0);

#pragma unroll
  for (int i = 0; i < ITERS; ++i) {
    if (i + 1 < ITERS) {
      issue(i + 1);   // prefetch next plane into the other buffer
      WAIT_ASYNC(2);  // stage i's 2 ops done; next stage's 2 still in flight
    } else {
      WAIT_ASYNC(0);
    }

    const f4 r0 = *(const f4*)&lds[i & 1][0][tid * 4];
    const f4 r1 = *(const f4*)&lds[i & 1][1][tid * 4];

    const float s0 = (r0.x + r0.y) + (r1.x + r1.y);
    const float s1 = (r0.z + r0.w) + (r1.z + r1.w);
    const float q0 = (r0.x * r0.x + r0.y * r0.y) + (r1.x * r1.x + r1.y * r1.y);
    const float q1 = (r0.z * r0.z + r0.w * r0.w) + (r1.z * r1.z + r1.w * r1.w);

    f2 v;
    v.x = fastdiv(q0, s0);
    v.y = fastdiv(q1, s1);

    // i*OSTRIDE folds into the store's 24-bit immediate offset.
    __builtin_nontemporal_store(v, (f2*)(o + (size_t)i * OSTRIDE));
  }
}

// Generic fallback for leftover planes (never used for the 16x96 shape, but
// keeps the launcher correct for any plane count). Plain coalesced b128 loads.
__global__ void stoch_pool2x2_tail(const float* __restrict__ x,
                                   float* __restrict__ out,
                                   int planeStart, int nPairs) {
  const int idx = blockIdx.x * blockDim.x + threadIdx.x;
  if (idx >= nPairs) return;
  const int plane = planeStart + idx / QPP;
  const int q  = idx - (idx / QPP) * QPP;
  const int io = q / JG;
  const int jg = q - io * JG;
  const float* g0 = x + ((size_t)plane * W + 2 * io) * W + jg * 4;
  const f4 r0 = *(const f4*)g0;
  const f4 r1 = *(const f4*)(g0 + W);
  const float s0 = (r0.x + r0.y) + (r1.x + r1.y);
  const float s1 = (r0.z + r0.w) + (r1.z + r1.w);
  const float q0 = (r0.x * r0.x + r0.y * r0.y) + (r1.x * r1.x + r1.y * r1.y);
  const float q1 = (r0.z * r0.z + r0.w * r0.w) + (r1.z * r1.z + r1.w * r1.w);
  f2 v; v.x = fastdiv(q0, s0); v.y = fastdiv(q1, s1);
  *(f2*)(out + ((size_t)plane * Ho + io) * Ho + jg * 2) = v;
}

extern "C" void kernel_launch(void* const* d_in, const int* in_sizes, int n_in,
                              void* d_out, int out_size, void* d_ws, size_t ws_size,
                              hipStream_t stream) {
  const float* x = (const float*)d_in[0];
  float* out     = (float*)d_out;

  const long long total = (long long)in_sizes[0];
  const int planes = (int)(total / (long long)(W * W));   // 16*96 = 1536
  const int groups = planes / ITERS;                       // 192
  const int rem    = planes - groups * ITERS;              // 0 for this shape

  if (groups > 0) {
    dim3 grid(QPP / BLOCK, groups);   // (28, 192)
    stoch_pool2x2_async<<<grid, BLOCK, 0, stream>>>(x, out);
  }
  if (rem > 0) {
    const int nPairs = rem * QPP;
    const int tb = 256;
    stoch_pool2x2_tail<<<(nPairs + tb - 1) / tb, tb, 0, stream>>>(
        x, out, groups * ITERS, nPairs);
  }
}